// LMSELossFunc_40819369181888
// MI455X (gfx1250) — compile-verified
//
#include <hip/hip_runtime.h>
#include <hip/hip_bf16.h>
#include <stdint.h>

// ---------------------------------------------------------------------------
// LMSE (scale-invariant local MSE) loss for [16,3,512,512] f32 images.
// Per 20x20 window (stride 10): ssq = Scc - a^2/v (v>EPS), total = Scc,
// with Scc = sum(m c c), a = sum(m c e), v = sum(m e e).
// Output scalar = sum_{b,ch}(ssq/total) / 48.
// Bandwidth-bound: data movement via CDNA5 Tensor Data Mover (TDM) -> LDS.
// ---------------------------------------------------------------------------

typedef __attribute__((ext_vector_type(4))) unsigned int uint32x4_t_;
typedef __attribute__((ext_vector_type(8))) int          int32x8_t_;
typedef __attribute__((ext_vector_type(4))) int          int32x4_t_;

#define WIN    20
#define STR    10
#define IMG    512
#define NWX    50
#define NWY    50
#define BATCH  16
#define CHANS  3
#define NIMG   (BATCH * CHANS)   // 48
#define EPSV   1e-5f

// dynamic LDS layout (bytes)
#define OFF_C   0
#define OFF_E   40960
#define OFF_M   81920
#define OFF_CC  122880
#define OFF_CE  124928
#define OFF_EE  126976
#define OFF_BLK 129024
#define SMEM_BYTES (OFF_BLK + 64)

// Issue one TDM 2-D tile load (global -> LDS).  D# per CDNA5 ISA:
// group0: [1:0]=count=1, [63:32]=lds_addr, [120:64]=global_addr, [127:126]=type=2
// group1: data_size=2(4B), tensor_dim0/1, tile_dim0/1, tensor_dim0_stride
__device__ __forceinline__ void tdm_load_2d(unsigned lds_addr, const float* gptr,
                                            unsigned tensor_w, unsigned tensor_h,
                                            unsigned tile_w, unsigned tile_h,
                                            unsigned row_stride_elems) {
    unsigned long long ga = (unsigned long long)(uintptr_t)gptr;
    uint32x4_t_ g0;
    g0[0] = 1u;                                              // count=1, user mode
    g0[1] = lds_addr;                                        // LDS byte address
    g0[2] = (unsigned)(ga & 0xFFFFFFFFu);                    // global_addr[31:0]
    g0[3] = (unsigned)((ga >> 32) & 0x1FFFFFFu) | (2u << 30);// global_addr[56:32] | type=2

    int32x8_t_ g1;
    g1[0] = (int)(2u << 16);                                 // wg_mask=0, data_size=4B
    g1[1] = (int)((tensor_w & 0xFFFFu) << 16);               // atomic_bar=0 | tensor_dim0 lo
    g1[2] = (int)(((tensor_w >> 16) & 0xFFFFu) |
                  ((tensor_h & 0xFFFFu) << 16));             // tensor_dim0 hi | tensor_dim1 lo
    g1[3] = (int)(((tensor_h >> 16) & 0xFFFFu) |
                  ((tile_w & 0xFFFFu) << 16));               // tensor_dim1 hi | tile_dim0
    g1[4] = (int)(tile_h & 0xFFFFu);                         // tile_dim1, tile_dim2=0
    g1[5] = (int)row_stride_elems;                           // tensor_dim0_stride[31:0]
    g1[6] = 0;                                               // stride hi / dim1_stride lo
    g1[7] = 0;

    int32x4_t_ z4 = {0, 0, 0, 0};
    int32x8_t_ z8 = {0, 0, 0, 0, 0, 0, 0, 0};
    // clang-23 / therock-10.0 6-arg form: (g0, g1, g2, g3, g4, cpol)
    __builtin_amdgcn_tensor_load_to_lds(g0, g1, z4, z4, z8, 0);
}

__global__ void lmse_init_kernel(float* __restrict__ acc) {
    int t = blockIdx.x * blockDim.x + threadIdx.x;
    if (t < NIMG * 2) acc[t] = 0.0f;
}

__global__ __launch_bounds__(256) void lmse_main_kernel(const float* __restrict__ cor,
                                                        const float* __restrict__ est,
                                                        const float* __restrict__ msk,
                                                        float* __restrict__ acc) {
    extern __shared__ char smem[];
    float* sh_c   = (float*)(smem + OFF_C);
    float* sh_e   = (float*)(smem + OFF_E);
    float* sh_m   = (float*)(smem + OFF_M);
    float* col_cc = (float*)(smem + OFF_CC);
    float* col_ce = (float*)(smem + OFF_CE);
    float* col_ee = (float*)(smem + OFF_EE);
    float* blk    = (float*)(smem + OFF_BLK);  // [2] = {ssq, total}

    const int bid = blockIdx.x;          // 0..2399
    const int wy  = bid % NWY;           // window row
    const int img = bid / NWY;           // b*3+ch
    const int r0  = wy * STR;            // first pixel row of strip
    const size_t base = ((size_t)img * IMG + (size_t)r0) * IMG;

    if (threadIdx.x == 0) { blk[0] = 0.0f; blk[1] = 0.0f; }

    // Wave 0 issues three async 2-D TDM tile loads (20x512 f32 strips) into LDS.
    if (threadIdx.x < 32) {
        tdm_load_2d((unsigned)(uintptr_t)sh_c, cor + base, IMG, WIN, IMG, WIN, IMG);
        tdm_load_2d((unsigned)(uintptr_t)sh_e, est + base, IMG, WIN, IMG, WIN, IMG);
        tdm_load_2d((unsigned)(uintptr_t)sh_m, msk + base, IMG, WIN, IMG, WIN, IMG);
        __builtin_amdgcn_s_wait_tensorcnt(0);
    }
    __syncthreads();

    // Phase 1: per-column sums over the 20 strip rows (removes x-overlap redundancy).
    for (int x = threadIdx.x; x < IMG; x += 256) {
        float cc = 0.0f, ce = 0.0f, ee = 0.0f;
#pragma unroll
        for (int r = 0; r < WIN; ++r) {
            float cv = sh_c[r * IMG + x];
            float ev = sh_e[r * IMG + x];
            float mv = sh_m[r * IMG + x];
            float mc = mv * cv;
            float me = mv * ev;
            cc = fmaf(mc, cv, cc);
            ce = fmaf(mc, ev, ce);
            ee = fmaf(me, ev, ee);
        }
        col_cc[x] = cc; col_ce[x] = ce; col_ee[x] = ee;
    }
    __syncthreads();

    // Phase 2: 50 windows in this row; window wx sums 20 consecutive colsums.
    if (threadIdx.x < NWX) {
        const int x0 = threadIdx.x * STR;
        float scc = 0.0f, a = 0.0f, v = 0.0f;
#pragma unroll
        for (int k = 0; k < WIN; ++k) {
            scc += col_cc[x0 + k];
            a   += col_ce[x0 + k];
            v   += col_ee[x0 + k];
        }
        // ssq_w = Scc - 2*alpha*a + alpha^2*v ; alpha = a/v if v>EPS else 0
        float ssq = scc - ((v > EPSV) ? (a * a / v) : 0.0f);
        atomicAdd(&blk[0], ssq);   // ds_add_f32
        atomicAdd(&blk[1], scc);
    }
    __syncthreads();

    if (threadIdx.x == 0) {
        atomicAdd(&acc[img * 2 + 0], blk[0]);   // global_atomic_add_f32
        atomicAdd(&acc[img * 2 + 1], blk[1]);
    }
}

__global__ void lmse_final_kernel(const float* __restrict__ acc, float* __restrict__ out) {
    int t = threadIdx.x;              // single wave32
    float r = 0.0f;
    for (int i = t; i < NIMG; i += 32)
        r += acc[i * 2 + 0] / acc[i * 2 + 1];
#pragma unroll
    for (int off = 16; off > 0; off >>= 1)
        r += __shfl_xor(r, off, 32);
    if (t == 0) out[0] = r / (float)NIMG;   // mean over batch of (sum_ch ratio) / 3
}

extern "C" void kernel_launch(void* const* d_in, const int* in_sizes, int n_in,
                              void* d_out, int out_size, void* d_ws, size_t ws_size,
                              hipStream_t stream) {
    (void)in_sizes; (void)n_in; (void)out_size; (void)ws_size;
    const float* cor = (const float*)d_in[0];
    const float* est = (const float*)d_in[1];
    const float* msk = (const float*)d_in[2];
    float* acc = (float*)d_ws;      // 96 floats: {ssq,total} per (b,ch)
    float* out = (float*)d_out;

    lmse_init_kernel<<<1, 128, 0, stream>>>(acc);
    lmse_main_kernel<<<NIMG * NWY, 256, SMEM_BYTES, stream>>>(cor, est, msk, acc);
    lmse_final_kernel<<<1, 32, 0, stream>>>(acc, out);
}